// MossAudioTokenizerResidualVQ_46849503264985
// MI455X (gfx1250) — compile-verified
//
#include <hip/hip_runtime.h>
#include <hip/hip_bf16.h>

// ---------------------------------------------------------------------------
// MossAudioTokenizerResidualVQ — restructured RVQ for gfx1250 (wave32, WMMA)
//
//   E_q   = in_w_q @ (z*mask)                                (K1, WMMA f32 GEMM)
//   z_e^q = E_q + in_b_q - mask*sum_{p<q}(M[q,p] zq_p + in_w_q ob_p)
//   idx   = argmax_k ( z_e . cb_k - 0.5|cb_k|^2 )            (K2, WMMA scan,
//                                                             codebook staged to
//                                                             LDS via TDM)
//   qout  = mask*( sum_q out_w_q zq_q + sum_q ob_q )         (K3, WMMA GEMM)
// ---------------------------------------------------------------------------

typedef float v2f __attribute__((ext_vector_type(2)));
typedef float v8f __attribute__((ext_vector_type(8)));
typedef unsigned int u32x4 __attribute__((ext_vector_type(4)));
typedef int i32x4 __attribute__((ext_vector_type(4)));
typedef int i32x8 __attribute__((ext_vector_type(8)));

static constexpr int NQ  = 32;
static constexpr int KC  = 1024;   // codebook size
static constexpr int CD  = 8;      // codebook dim
static constexpr int DD  = 1024;   // rvq dim
static constexpr int BB  = 8;
static constexpr int TT  = 4096;
static constexpr int BT  = BB * TT;          // 32768 positions
static constexpr int ROWS = NQ * CD;         // 256 stacked encoder rows
static constexpr int CBPAK = 10240;          // floats per packed codebook block
                                             // (2048 float4 A-frags + 2048 bias)

#if defined(__has_builtin)
#if __has_builtin(__builtin_amdgcn_tensor_load_to_lds) && \
    __has_builtin(__builtin_amdgcn_s_wait_tensorcnt)
#define USE_TDM 1
#endif
#endif

__device__ __forceinline__ v8f wmma_f32_4(v2f a, v2f b, v8f c) {
  // D = A(16x4) * B(4x16) + C(16x16), fp32
  return __builtin_amdgcn_wmma_f32_16x16x4_f32(
      /*neg_a=*/false, a, /*neg_b=*/false, b,
      /*c_mod=*/(short)0, c, /*reuse_a=*/false, /*reuse_b=*/false);
}

#ifdef USE_TDM
// One-row 2D TDM descriptor: copy nElem consecutive f32 from gsrc to LDS.
// 6-arg builtin form: (u32x4 g0, i32x8 g1, i32x4 g2, i32x4 g3, i32x8 pad, cpol)
__device__ __forceinline__ void tdm_copy_f32(unsigned lds_addr, const void* gsrc,
                                             int nElem) {
  unsigned long long ga = (unsigned long long)(__SIZE_TYPE__)gsrc;
  u32x4 g0;
  g0[0] = 1u;                                           // count=1 (valid D#)
  g0[1] = lds_addr;                                     // lds_addr [63:32]
  g0[2] = (unsigned)(ga & 0xFFFFFFFFu);                 // global_addr lo
  g0[3] = (unsigned)((ga >> 32) & 0x1FFFFFFu)           // global_addr [56:32]
          | 0x80000000u;                                // type=2 ("image")
  i32x8 g1;
  g1[0] = 0x20000;                                      // data_size=2 (4B)
  g1[1] = (nElem & 0xFFFF) << 16;                       // tensor_dim0 lo16
  g1[2] = ((nElem >> 16) & 0xFFFF) | (1 << 16);         // tensor_dim0 hi16, dim1=1
  g1[3] = (nElem & 0xFFFF) << 16;                       // tile_dim0
  g1[4] = 1;                                            // tile_dim1=1
  g1[5] = nElem;                                        // tensor_dim0_stride lo32
  g1[6] = 0;
  g1[7] = 0;
  i32x4 z4 = {0, 0, 0, 0};
  i32x8 z8 = {0, 0, 0, 0, 0, 0, 0, 0};
  __builtin_amdgcn_tensor_load_to_lds(g0, g1, z4, z4, z8, 0);
}
#endif

// ---------------------------------------------------------------------------
// K0a: misc small precomputes: obsum[d], vsum[q][c], copy input_length
// ---------------------------------------------------------------------------
__global__ void k0_misc(const float* __restrict__ in_w,
                        const float* __restrict__ ob,
                        const int*   __restrict__ len,
                        float* __restrict__ obsum,
                        float* __restrict__ vsum,
                        int*   __restrict__ len_out) {
  int tid = blockIdx.x * blockDim.x + threadIdx.x;    // 1024 threads total
  if (tid < DD) {
    float s = 0.f;
    for (int q = 0; q < NQ; ++q) s += ob[q * DD + tid];
    obsum[tid] = s;
  }
  if (tid < BB) len_out[tid] = len[tid];
  if (tid < ROWS) {                                    // (q,c)
    int q = tid >> 3, c = tid & 7;
    float s = 0.f;
    for (int p = 0; p < q; ++p)
      for (int d = 0; d < DD; ++d)
        s += in_w[(q * CD + c) * DD + d] * ob[p * DD + d];
    vsum[tid] = s;
  }
}

// ---------------------------------------------------------------------------
// K0b: M[q][p][c][j] = sum_d in_w[q,c,d] * out_w[p,d,j]    (32x32 8x8 mats)
// ---------------------------------------------------------------------------
__global__ void k0_mmat(const float* __restrict__ in_w,
                        const float* __restrict__ out_w,
                        float* __restrict__ Mmat) {
  int q = blockIdx.x >> 5, p = blockIdx.x & 31;
  int c = threadIdx.x >> 3, j = threadIdx.x & 7;
  float s = 0.f;
  for (int d = 0; d < DD; ++d)
    s += in_w[(q * CD + c) * DD + d] * out_w[(p * DD + d) * CD + j];
  Mmat[((q * NQ + p) * CD + c) * CD + j] = s;
}

// ---------------------------------------------------------------------------
// K0c: pack in_w (256x1024 row-major) into f32-WMMA A-fragment order
// fragment (rt,kc,lane): rows rt*16+(lane%16), K = kc*8+2h + {0,1,4,5}
// ---------------------------------------------------------------------------
__global__ void k0_pack_w1(const float* __restrict__ W, float4* __restrict__ out) {
  int gid = blockIdx.x * blockDim.x + threadIdx.x;   // 16*128*32 = 65536
  int lane = gid & 31, kc = (gid >> 5) & 127, rt = gid >> 12;
  int h = lane >> 4, n = lane & 15;
  const float* row = W + (size_t)(rt * 16 + n) * DD + (kc * 8 + 2 * h);
  out[gid] = make_float4(row[0], row[1], row[4], row[5]);
}

// K0d: pack W2[d][q*8+c] = out_w[q][d][c]  (1024x256) into A-fragment order
__global__ void k0_pack_w2(const float* __restrict__ out_w, float4* __restrict__ out) {
  int gid = blockIdx.x * blockDim.x + threadIdx.x;   // 64*32*32 = 65536
  int lane = gid & 31, kc = (gid >> 5) & 31, dt = gid >> 10;
  int h = lane >> 4, n = lane & 15;
  int d = dt * 16 + n;
  int k = kc * 8 + 2 * h;
  float4 f;
  f.x = out_w[(((k + 0) >> 3) * DD + d) * CD + ((k + 0) & 7)];
  f.y = out_w[(((k + 1) >> 3) * DD + d) * CD + ((k + 1) & 7)];
  f.z = out_w[(((k + 4) >> 3) * DD + d) * CD + ((k + 4) & 7)];
  f.w = out_w[(((k + 5) >> 3) * DD + d) * CD + ((k + 5) & 7)];
  out[gid] = f;
}

// K0e: pack codebooks per q into ONE contiguous 40KB block (TDM-friendly):
//   floats [0..8191]     : 2048 float4 A-fragments (16 codes x K=8 tiles)
//   floats [8192..10239] : bias column  -0.5|cb|^2 (0 for lane-half h=1)
__global__ void k0_pack_cb(const float* __restrict__ cb,
                           float* __restrict__ cbPak) {
  int gid = blockIdx.x * blockDim.x + threadIdx.x;   // 32*64*32 = 65536
  int lane = gid & 31, tt = (gid >> 5) & 63, q = gid >> 11;
  int h = lane >> 4, n = lane & 15;
  const float* row = cb + ((size_t)q * KC + tt * 16 + n) * CD;
  float* base = cbPak + (size_t)q * CBPAK;
  ((float4*)base)[tt * 32 + lane] =
      make_float4(row[2 * h], row[2 * h + 1], row[2 * h + 4], row[2 * h + 5]);
  float b = 0.f;
  if (h == 0) {
    float s = 0.f;
#pragma unroll
    for (int c = 0; c < CD; ++c) s += row[c] * row[c];
    b = -0.5f * s;
  }
  base[8192 + tt * 32 + lane] = b;
}

// ---------------------------------------------------------------------------
// K1: E[row][pos] = sum_d W1[row][d] * z[b][d][t]   (256x1024)x(1024x32768)
// one wave per (row-tile, 16-position tile); K streamed in chunks of 8.
// ---------------------------------------------------------------------------
__global__ __launch_bounds__(32) void k1_encode(const float*  __restrict__ z,
                                                const float4* __restrict__ w1sw,
                                                float* __restrict__ E) {
  int pt   = blockIdx.x;          // 0..BT/16-1
  int rt   = blockIdx.y;          // 0..15
  int lane = threadIdx.x;
  int h = lane >> 4, n = lane & 15;
  int b = pt / (TT / 16);
  int t = (pt % (TT / 16)) * 16 + n;
  const float*  zb = z + (size_t)b * DD * TT + t;
  const float4* A  = w1sw + (size_t)rt * 128 * 32 + lane;
  v8f acc = {};
  for (int kc = 0; kc < 128; ++kc) {
    float4 a = A[(size_t)kc * 32];
    int k0 = kc * 8 + h;
    v2f a01 = {a.x, a.y}, a23 = {a.z, a.w};
    v2f b01 = { zb[(size_t)(k0 + 0) * TT], zb[(size_t)(k0 + 2) * TT] };
    v2f b23 = { zb[(size_t)(k0 + 4) * TT], zb[(size_t)(k0 + 6) * TT] };
    acc = wmma_f32_4(a01, b01, acc);
    acc = wmma_f32_4(a23, b23, acc);
  }
  float* Ecol = E + (size_t)pt * 16 + n;
#pragma unroll
  for (int r = 0; r < 8; ++r)
    Ecol[(size_t)(rt * 16 + r + 8 * h) * BT] = acc[r];
}

// ---------------------------------------------------------------------------
// K2: sequential quantization in C=8 space.  256 threads = 256 positions.
// Per quantizer: TDM stages the 40KB packed codebook block into LDS (overlapped
// with the VALU history-correction phase), then WMMA code scan from LDS
// (codes as A, encodings as B, -0.5|cb|^2 folded via a 3rd WMMA with B=1).
// ---------------------------------------------------------------------------
__global__ __launch_bounds__(256) void k2_quantize(
    const float*  __restrict__ E,     const float* __restrict__ inb,
    const float*  __restrict__ Mmat,  const float* __restrict__ vsum,
    const float*  __restrict__ cbPak, const float* __restrict__ cb,
    const int*    __restrict__ len,
    float* __restrict__ zqh,          int* __restrict__ idx_out) {
  __shared__ __align__(16) float lds_cb[CBPAK];   // 40 KB staged codebook
  __shared__ float encS[CD * 256];                //  8 KB enc transpose

  int tid  = threadIdx.x, lane = tid & 31, w = tid >> 5;
  int pos  = blockIdx.x * 256 + tid;
  int b    = pos >> 12, t = pos & (TT - 1);
  bool valid = t < len[b];
  int h = lane >> 4, n = lane & 15;
  int wavebase = blockIdx.x * 256 + w * 32;

  for (int q = 0; q < NQ; ++q) {
    const float* gblk = cbPak + (size_t)q * CBPAK;
#ifdef USE_TDM
    if (tid == 0)   // one wave issues the DMA; overlaps with corrections below
      tdm_copy_f32((unsigned)(__SIZE_TYPE__)&lds_cb[0], gblk, CBPAK);
#else
    for (int i = tid; i < CBPAK / 4; i += 256)
      ((float4*)lds_cb)[i] = ((const float4*)gblk)[i];
#endif

    // ---- per-thread encoding with history correction -----------------
    float e[CD];
#pragma unroll
    for (int c = 0; c < CD; ++c) {
      float v = inb[q * CD + c];
      if (valid) v += E[(size_t)(q * CD + c) * BT + pos] - vsum[q * CD + c];
      e[c] = v;
    }
    const float* Mq = Mmat + (size_t)q * NQ * 64;
    for (int p = 0; p < q; ++p) {
      float zq[CD];
#pragma unroll
      for (int c = 0; c < CD; ++c)
        zq[c] = zqh[(size_t)(p * CD + c) * BT + pos];
      const float* Mp = Mq + p * 64;
#pragma unroll
      for (int c = 0; c < CD; ++c) {
        float s = e[c];
#pragma unroll
        for (int j = 0; j < CD; ++j) s -= Mp[c * 8 + j] * zq[j];
        e[c] = s;
      }
    }
    // ---- transpose encodings into B-fragment order via LDS -----------
#pragma unroll
    for (int c = 0; c < CD; ++c) encS[c * 256 + tid] = e[c];

#ifdef USE_TDM
    if (tid == 0) __builtin_amdgcn_s_wait_tensorcnt(0);
#endif
    __syncthreads();   // codebook staged + encS visible

    float bestVal[2]; int bestIdx[2];
#pragma unroll
    for (int g = 0; g < 2; ++g) {
      int col = w * 32 + g * 16 + n;
      v2f b01 = { encS[(0 + h) * 256 + col], encS[(2 + h) * 256 + col] };
      v2f b23 = { encS[(4 + h) * 256 + col], encS[(6 + h) * 256 + col] };
      v2f bB  = { h ? 0.f : 1.f, 0.f };
      float bv = -3.402823e38f; int bi = 0;
      const float4* Aq = (const float4*)lds_cb + lane;
      const float*  Bq = lds_cb + 8192 + lane;
      for (int tt = 0; tt < 64; ++tt) {
        float4 a = Aq[tt * 32];                 // ds_load_b128, conflict-free
        v2f a01 = {a.x, a.y}, a23 = {a.z, a.w};
        v2f aB  = { Bq[tt * 32], 0.f };
        v8f d = {};
        d = wmma_f32_4(a01, b01, d);
        d = wmma_f32_4(a23, b23, d);
        d = wmma_f32_4(aB,  bB,  d);     // folds -0.5|cb|^2 per code row
        int codebase = tt * 16 + 8 * h;
#pragma unroll
        for (int r = 0; r < 8; ++r) {
          float s = d[r];
          bool gt = s > bv;              // strict > keeps lowest index on tie
          bv = gt ? s : bv;
          bi = gt ? (codebase + r) : bi;
        }
      }
      bestVal[g] = bv; bestIdx[g] = bi;
    }
    __syncthreads();

    // ---- combine the two lane-halves, write idx + zq history ---------
#pragma unroll
    for (int g = 0; g < 2; ++g) {
      float ov = __shfl_xor(bestVal[g], 16, 32);
      int   oi = __shfl_xor(bestIdx[g], 16, 32);
      bool take = (ov > bestVal[g]) || (ov == bestVal[g] && oi < bestIdx[g]);
      float fv = take ? ov : bestVal[g]; (void)fv;
      int   fi = take ? oi : bestIdx[g];
      if (lane < 16) {
        int posW = wavebase + g * 16 + lane;
        int bW = posW >> 12, tW = posW & (TT - 1);
        bool vW = tW < len[bW];
        idx_out[(size_t)q * BT + posW] = fi;
        const float* row = cb + ((size_t)q * KC + fi) * CD;
#pragma unroll
        for (int c = 0; c < CD; ++c)
          zqh[(size_t)(q * CD + c) * BT + posW] = vW ? row[c] : 0.f;
      }
    }
    __threadfence_block();
    __syncthreads();
  }
}

// ---------------------------------------------------------------------------
// K3: qout[b][d][t] = mask * ( sum_{q,c} W2[d][qc] * cb[q][idx][c] + obsum[d] )
// 256 threads, 32 positions per block, full D; X tile gathered into LDS.
// ---------------------------------------------------------------------------
__global__ __launch_bounds__(256) void k3_decode(
    const float4* __restrict__ w2sw, const float* __restrict__ cb,
    const int* __restrict__ idx_in,  const float* __restrict__ obsum,
    const int* __restrict__ len,     float* __restrict__ qout) {
  __shared__ float X[ROWS * 32];   // 32 KB

  int tid = threadIdx.x, lane = tid & 31, w = tid >> 5;
  int pos0 = blockIdx.x * 32;
  int j = tid & 31, qg = tid >> 5;
#pragma unroll
  for (int qq = 0; qq < 4; ++qq) {
    int q = qg * 4 + qq;
    int id = idx_in[(size_t)q * BT + pos0 + j];
    const float* row = cb + ((size_t)q * KC + id) * CD;
#pragma unroll
    for (int c = 0; c < CD; ++c) X[(q * CD + c) * 32 + j] = row[c];
  }
  __syncthreads();

  int h = lane >> 4, n = lane & 15;
  for (int dt = w; dt < 64; dt += 8) {
    v8f acc0 = {}, acc1 = {};
    const float4* A = w2sw + (size_t)dt * 32 * 32 + lane;
    for (int kc = 0; kc < 32; ++kc) {
      float4 a = A[(size_t)kc * 32];
      v2f a01 = {a.x, a.y}, a23 = {a.z, a.w};
      int k0 = kc * 8 + h;
      v2f b01a = { X[(k0 + 0) * 32 + n],      X[(k0 + 2) * 32 + n] };
      v2f b23a = { X[(k0 + 4) * 32 + n],      X[(k0 + 6) * 32 + n] };
      acc0 = wmma_f32_4(a01, b01a, acc0);
      acc0 = wmma_f32_4(a23, b23a, acc0);
      v2f b01b = { X[(k0 + 0) * 32 + 16 + n], X[(k0 + 2) * 32 + 16 + n] };
      v2f b23b = { X[(k0 + 4) * 32 + 16 + n], X[(k0 + 6) * 32 + 16 + n] };
      acc1 = wmma_f32_4(a01, b01b, acc1);
      acc1 = wmma_f32_4(a23, b23b, acc1);
    }
#pragma unroll
    for (int g = 0; g < 2; ++g) {
      v8f acc = g ? acc1 : acc0;
      int pos = pos0 + g * 16 + n;
      int b = pos >> 12, t = pos & (TT - 1);
      bool v = t < len[b];
      float* out = qout + (size_t)b * DD * TT + t;
#pragma unroll
      for (int r = 0; r < 8; ++r) {
        int d = dt * 16 + r + 8 * h;
        out[(size_t)d * TT] = v ? (acc[r] + obsum[d]) : 0.f;
      }
    }
  }
}

// ---------------------------------------------------------------------------
extern "C" void kernel_launch(void* const* d_in, const int* in_sizes, int n_in,
                              void* d_out, int out_size, void* d_ws, size_t ws_size,
                              hipStream_t stream) {
  const float* z     = (const float*)d_in[0];
  const float* in_w  = (const float*)d_in[1];
  const float* in_b  = (const float*)d_in[2];
  const float* out_w = (const float*)d_in[3];
  const float* out_b = (const float*)d_in[4];
  const float* cbs   = (const float*)d_in[5];
  const int*   len   = (const int*)d_in[6];

  float* qout    = (float*)d_out;                       // 8*1024*4096 f32
  int*   idx_out = (int*)((float*)d_out + (size_t)BB * DD * TT);  // 32*8*4096 i32
  int*   len_out = idx_out + (size_t)NQ * BT;           // 8 i32

  float* ws    = (float*)d_ws;
  float* E     = ws;                       ws += (size_t)ROWS * BT;   // 8.4M
  float* W1sw  = ws;                       ws += (size_t)ROWS * DD;   // 256K
  float* W2sw  = ws;                       ws += (size_t)DD * ROWS;   // 256K
  float* cbPak = ws;                       ws += (size_t)NQ * CBPAK;  // 320K
  float* Mmat  = ws;                       ws += (size_t)NQ * NQ * 64;
  float* vsum  = ws;                       ws += ROWS;
  float* obsum = ws;                       ws += DD;
  float* zqh   = ws;                       ws += (size_t)ROWS * BT;   // 8.4M

  k0_misc   <<<4, 256, 0, stream>>>(in_w, out_b, len, obsum, vsum, len_out);
  k0_mmat   <<<NQ * NQ, 64, 0, stream>>>(in_w, out_w, Mmat);
  k0_pack_w1<<<256, 256, 0, stream>>>(in_w, (float4*)W1sw);
  k0_pack_w2<<<256, 256, 0, stream>>>(out_w, (float4*)W2sw);
  k0_pack_cb<<<256, 256, 0, stream>>>(cbs, cbPak);

  k1_encode <<<dim3(BT / 16, 16), 32, 0, stream>>>(z, (const float4*)W1sw, E);
  k2_quantize<<<BT / 256, 256, 0, stream>>>(E, in_b, Mmat, vsum, cbPak,
                                            cbs, len, zqh, idx_out);
  k3_decode <<<BT / 32, 256, 0, stream>>>((const float4*)W2sw, cbs, idx_out,
                                          obsum, len, qout);
}